// LogicLayer_13657996002166
// MI455X (gfx1250) — compile-verified
//
#include <hip/hip_runtime.h>
#include <cstdint>

// Problem sizes (fixed by the reference).
#define IN_DIM   65536
#define OUT_DIM  65536
#define NBATCH   256
#define SPLIT    4                    // output chunks per batch row
#define LDS_BYTES (IN_DIM * 4)        // 256 KB row of x staged in LDS (<= 320 KB/WGP)

typedef __attribute__((ext_vector_type(2))) float v2f;   // f32 16x16x4 A/B frag: 2 VGPRs
typedef __attribute__((ext_vector_type(4))) float v4f;   // native 128-bit vector (NT store)
typedef __attribute__((ext_vector_type(8))) float v8f;   // f32 16x16 C/D frag: 8 VGPRs

// Coefficients of each logic fn in basis (1, a, b, a*b), flattened [f*4 + c].
// Compile-time constant: only ever indexed with literal indices so it folds
// into inline constants (no memory, no select chains).
constexpr float KT[64] = {
  0.f, 0.f, 0.f, 0.f,    // 0: false
  0.f, 0.f, 0.f, 1.f,    // 1: a AND b
  0.f, 1.f, 0.f,-1.f,    // 2: a AND NOT b
  0.f, 1.f, 0.f, 0.f,    // 3: a
  0.f, 0.f, 1.f,-1.f,    // 4: NOT a AND b
  0.f, 0.f, 1.f, 0.f,    // 5: b
  0.f, 1.f, 1.f,-2.f,    // 6: XOR
  0.f, 1.f, 1.f,-1.f,    // 7: OR
  1.f,-1.f,-1.f, 1.f,    // 8: NOR
  1.f,-1.f,-1.f, 2.f,    // 9: XNOR
  1.f, 0.f,-1.f, 0.f,    // 10: NOT b
  1.f, 0.f,-1.f, 1.f,    // 11: a OR NOT b
  1.f,-1.f, 0.f, 0.f,    // 12: NOT a
  1.f,-1.f, 0.f, 1.f,    // 13: NOT a OR b
  1.f, 0.f, 0.f,-1.f,    // 14: NAND
  1.f, 0.f, 0.f, 0.f,    // 15: true
};

// Select one of 4 compile-time constants by runtime cc (3 v_cndmask w/ inline consts).
__device__ __forceinline__ float ksel(float k0, float k1, float k2, float k3, int cc) {
  float r = (cc == 1) ? k1 : k0;
  r = (cc == 2) ? k2 : r;
  r = (cc == 3) ? k3 : r;
  return r;
}

// Preprocess: per-output fp32 softmax over 16 weights, then
// C[o,0:4] = softmax(W[o]) @ KT as 4 chained V_WMMA_F32_16X16X4_F32
// (full-precision matrix pipe), 16 output rows per wave.
// Also packs the two int32 gather indices per output into an int2.
__global__ __launch_bounds__(256)
void coef_pack_kernel(const float* __restrict__ W,
                      const int*  __restrict__ IND,     // [2][OUT_DIM] int32
                      float* __restrict__ Ccoef,        // [OUT_DIM*4]
                      int2*  __restrict__ IDX)          // [OUT_DIM]
{
  const int lane = threadIdx.x & 31;
  const int wave = threadIdx.x >> 5;
  const int o0   = (blockIdx.x * 8 + wave) * 16;   // 16-row tile per wave
  const int hi   = lane >> 4;                      // lane group (K split per ISA layout)
  const int r    = o0 + (lane & 15);               // this lane's weight row (M index)

  // fp32 softmax of weights row r (both lane groups need the full row).
  float p[16];
  const float* wr = W + (size_t)r * 16;
  #pragma unroll
  for (int k = 0; k < 16; ++k) p[k] = wr[k];
  float m = p[0];
  #pragma unroll
  for (int k = 1; k < 16; ++k) m = fmaxf(m, p[k]);
  float s = 0.f;
  #pragma unroll
  for (int k = 0; k < 16; ++k) { p[k] = __expf(p[k] - m); s += p[k]; }
  const float inv = 1.f / s;
  #pragma unroll
  for (int k = 0; k < 16; ++k) p[k] *= inv;

  // D(16x16) = sum_t A_t(16x4) * B_t(4x16); only columns 0..3 of D are used.
  // A 16x4 f32 layout (ISA 7.12.2): lanes0-15 v0=K0,v1=K1 ; lanes16-31 v0=K2,v1=K3.
  // B 4x16 mirrors it:             lanes0-15 v0=row0,v1=row1 ; lanes16-31 v0=row2,v1=row3.
  const int col = lane & 15;
  const int cc  = col & 3;
  const bool act = col < 4;
  v8f d = {0.f, 0.f, 0.f, 0.f, 0.f, 0.f, 0.f, 0.f};

#define WSTEP(t)                                                               \
  {                                                                            \
    v2f a, b;                                                                  \
    a[0] = hi ? p[4*(t) + 2] : p[4*(t) + 0];                                   \
    a[1] = hi ? p[4*(t) + 3] : p[4*(t) + 1];                                   \
    const float b0lo = ksel(KT[(4*(t)+0)*4+0], KT[(4*(t)+0)*4+1],              \
                            KT[(4*(t)+0)*4+2], KT[(4*(t)+0)*4+3], cc);         \
    const float b0hi = ksel(KT[(4*(t)+2)*4+0], KT[(4*(t)+2)*4+1],              \
                            KT[(4*(t)+2)*4+2], KT[(4*(t)+2)*4+3], cc);         \
    const float b1lo = ksel(KT[(4*(t)+1)*4+0], KT[(4*(t)+1)*4+1],              \
                            KT[(4*(t)+1)*4+2], KT[(4*(t)+1)*4+3], cc);         \
    const float b1hi = ksel(KT[(4*(t)+3)*4+0], KT[(4*(t)+3)*4+1],              \
                            KT[(4*(t)+3)*4+2], KT[(4*(t)+3)*4+3], cc);         \
    b[0] = act ? (hi ? b0hi : b0lo) : 0.f;                                     \
    b[1] = act ? (hi ? b1hi : b1lo) : 0.f;                                     \
    d = __builtin_amdgcn_wmma_f32_16x16x4_f32(false, a, false, b,              \
                                              (short)0, d, false, false);      \
  }

  WSTEP(0)
  WSTEP(1)
  WSTEP(2)
  WSTEP(3)
#undef WSTEP

  // D readout: VGPR j holds (M=j, N=lane) for lanes<16, (M=j+8, N=lane-16) for lanes>=16.
  if (act) {
    const int mo = hi ? 8 : 0;
    #pragma unroll
    for (int j = 0; j < 8; ++j)
      Ccoef[(size_t)(o0 + mo + j) * 4 + col] = d[j];
  }

  // Pack gather indices (one per row, lanes 0..15 of each wave).
  if (lane < 16) {
    int2 pr;
    pr.x = IND[r];
    pr.y = IND[OUT_DIM + r];
    IDX[r] = pr;
  }
}

// Main kernel: one workgroup owns (batch, output-chunk). The batch's full x row
// (256 KB) is DMA'd into LDS via the CDNA5 async-to-LDS path, then all 2*16.7M
// random gathers hit LDS banks instead of divergent VMEM. Output is streamed
// with non-temporal b128 stores so x/IDX/C4 stay resident in the 192 MB L2.
__global__ __launch_bounds__(1024)
void logic_main_kernel(const float*  __restrict__ x,
                       const float4* __restrict__ C4,
                       const int2*   __restrict__ IDX,
                       float*        __restrict__ out)
{
  extern __shared__ float lds[];                // dynamic LDS at offset 0
  const int tid   = threadIdx.x;
  const int batch = blockIdx.x;
  const float* __restrict__ row = x + (size_t)batch * IN_DIM;

  // Stage row: 1024 threads x 16 async b128 transfers = 65536 floats.
  #pragma unroll
  for (int i = 0; i < IN_DIM / (1024 * 4); ++i) {
    const int v4 = i * 1024 + tid;                              // float4 index
    const unsigned ldsOff = (unsigned)v4 * 16u;                 // LDS byte offset
    const unsigned long long gaddr =
        (unsigned long long)(uintptr_t)(row + (size_t)v4 * 4);
    asm volatile("global_load_async_to_lds_b128 %0, %1, off"
                 :: "v"(ldsOff), "v"(gaddr)
                 : "memory");
  }
  asm volatile("s_wait_asynccnt 0" ::: "memory");
  __syncthreads();

  const int chunk = OUT_DIM / SPLIT;            // 16384 outputs per WG
  const int obase = blockIdx.y * chunk;
  float* __restrict__ orow = out + (size_t)batch * OUT_DIM;

  // 4 outputs per thread per iteration: b128 index load, 4 coeff loads,
  // 8 LDS gathers, FMAs, one non-temporal b128 store.
  for (int q = tid * 4; q < chunk; q += 1024 * 4) {
    const int o = obase + q;
    const int4* ip = reinterpret_cast<const int4*>(IDX + o);
    const int4 i01 = ip[0];                     // (idx0,idx1) for o, o+1
    const int4 i23 = ip[1];                     // (idx0,idx1) for o+2, o+3
    const float4 c0 = C4[o + 0];
    const float4 c1 = C4[o + 1];
    const float4 c2 = C4[o + 2];
    const float4 c3 = C4[o + 3];

    const float a0 = lds[i01.x], b0 = lds[i01.y];
    const float a1 = lds[i01.z], b1 = lds[i01.w];
    const float a2 = lds[i23.x], b2 = lds[i23.y];
    const float a3 = lds[i23.z], b3 = lds[i23.w];

    v4f rv;
    rv.x = c0.x + c0.y * a0 + c0.z * b0 + c0.w * (a0 * b0);
    rv.y = c1.x + c1.y * a1 + c1.z * b1 + c1.w * (a1 * b1);
    rv.z = c2.x + c2.y * a2 + c2.z * b2 + c2.w * (a2 * b2);
    rv.w = c3.x + c3.y * a3 + c3.z * b3 + c3.w * (a3 * b3);
    __builtin_nontemporal_store(rv, reinterpret_cast<v4f*>(orow + o));
  }
}

extern "C" void kernel_launch(void* const* d_in, const int* in_sizes, int n_in,
                              void* d_out, int out_size, void* d_ws, size_t ws_size,
                              hipStream_t stream) {
  (void)in_sizes; (void)n_in; (void)out_size; (void)ws_size;

  const float* x   = (const float*)d_in[0];                 // [256, 65536] f32
  const float* w   = (const float*)d_in[1];                 // [65536, 16] f32
  const int*   ind = (const int*)d_in[2];                   // [2, 65536] int
  float*       out = (float*)d_out;                         // [256, 65536] f32

  float* Ccoef = (float*)d_ws;                              // 65536 * float4 = 1 MB
  int2*  IDX   = (int2*)((char*)d_ws + (size_t)OUT_DIM * 4 * sizeof(float));

  // 512 blocks x 8 waves x 16 rows = 65536 outputs
  coef_pack_kernel<<<OUT_DIM / 128, 256, 0, stream>>>(w, ind, Ccoef, IDX);

  (void)hipFuncSetAttribute(reinterpret_cast<const void*>(logic_main_kernel),
                            hipFuncAttributeMaxDynamicSharedMemorySize, LDS_BYTES);
  dim3 grid(NBATCH, SPLIT);
  logic_main_kernel<<<grid, 1024, LDS_BYTES, stream>>>(
      x, (const float4*)Ccoef, IDX, out);
}